// SimpleCompressedAttention_78280073937160
// MI455X (gfx1250) — compile-verified
//
#include <hip/hip_runtime.h>

typedef __attribute__((ext_vector_type(2))) float v2f;
typedef __attribute__((ext_vector_type(4))) float v4f;
typedef __attribute__((ext_vector_type(8))) float v8f;

#define B_  8
#define L_  4096
#define H_  32
#define K_  128
#define C_  512
#define QC_ 1536
#define HK_ (H_ * K_)   // 4096

// ---------------------------------------------------------------------------
// Phase 1a: q[b, h*K+k] = hidden_q[b,:] . q_w[h*K+k,:] + q_b
// ---------------------------------------------------------------------------
__global__ void qproj_kernel(const float* __restrict__ hq, const float* __restrict__ qw,
                             const float* __restrict__ qb, float* __restrict__ q) {
    int idx = blockIdx.x * blockDim.x + threadIdx.x;   // 0 .. B*HK-1
    int b  = idx >> 12;
    int hk = idx & (HK_ - 1);
    const float* hrow = hq + (size_t)b * QC_;
    const float* wrow = qw + (size_t)hk * QC_;
    float acc = qb[hk];
#pragma unroll 4
    for (int j = 0; j < QC_; j += 4) {
        v4f hv = *(const v4f*)(hrow + j);
        v4f wv = *(const v4f*)(wrow + j);
        acc += hv.x * wv.x + hv.y * wv.y + hv.z * wv.z + hv.w * wv.w;
    }
    q[idx] = acc;
}

// ---------------------------------------------------------------------------
// Phase 1b: q_eff[b,h,c] = sum_k q[b,h,k] * k_w[h*K+k, c]   (absorb k_w into q)
// ---------------------------------------------------------------------------
__global__ void qeff_kernel(const float* __restrict__ q, const float* __restrict__ kw,
                            float* __restrict__ qeff) {
    int idx = blockIdx.x * blockDim.x + threadIdx.x;   // (b*H+h)*C + c
    int c  = idx & (C_ - 1);
    int bh = idx >> 9;
    int h  = bh & (H_ - 1);
    const float* qrow = q + (size_t)bh * K_;
    const float* wcol = kw + (size_t)h * K_ * C_ + c;
    float acc = 0.f;
#pragma unroll 4
    for (int k = 0; k < K_; ++k)
        acc += qrow[k] * wcol[(size_t)k * C_];
    qeff[idx] = acc;
}

// ---------------------------------------------------------------------------
// Phase 2: scores[b,h,l] = q_eff[b,h,:] . kv[b,l,:]  via V_WMMA_F32_16X16X4_F32
// One wave: 16 l's (n-tile) x all 32 heads (two 16-row m-tiles, shared B frag).
// Fragment layout (f32 16x16x4): lane = row index (m or n), cols = kb + 2*(lane>>4)+{0,1}
//   -> both A and B fragments are contiguous float2 loads from row-major [*,C] arrays.
// ---------------------------------------------------------------------------
__global__ void scores_kernel(const float* __restrict__ qeff, const float* __restrict__ kv,
                              float* __restrict__ scores) {
    int wave  = (blockIdx.x * blockDim.x + threadIdx.x) >> 5;  // 0..2047
    int lane  = threadIdx.x & 31;
    int b     = wave >> 8;          // 256 n-tiles per batch
    int lbase = (wave & 255) * 16;
    int m     = lane & 15;
    int khalf = lane >> 4;

    const float* a0row = qeff + (size_t)(b * H_ + m)      * C_;
    const float* a1row = qeff + (size_t)(b * H_ + 16 + m) * C_;
    const float* brow  = kv   + (size_t)(b * L_ + lbase + m) * C_;

    v8f d0 = {}; v8f d1 = {};
    for (int kb = 0; kb < C_; kb += 4) {
        int col = kb + 2 * khalf;
        v2f a0 = *(const v2f*)(a0row + col);
        v2f a1 = *(const v2f*)(a1row + col);
        v2f bb = *(const v2f*)(brow  + col);
        d0 = __builtin_amdgcn_wmma_f32_16x16x4_f32(false, a0, false, bb, (short)0, d0, false, false);
        d1 = __builtin_amdgcn_wmma_f32_16x16x4_f32(false, a1, false, bb, (short)0, d1, false, false);
    }

    // D layout: vgpr r, lanes 0-15 -> M=r ; lanes 16-31 -> M=r+8 ; N = lane&15
    float* srow = scores + (size_t)(b * H_) * L_ + lbase + m;
#pragma unroll
    for (int r = 0; r < 8; ++r) {
        int row0 = khalf * 8 + r;
        srow[(size_t)row0        * L_] = d0[r];
        srow[(size_t)(row0 + 16) * L_] = d1[r];
    }
}

// ---------------------------------------------------------------------------
// Phase 3: in-place softmax over each scores row (length L)
// ---------------------------------------------------------------------------
__global__ void softmax_kernel(float* __restrict__ scores) {
    float* s = scores + (size_t)blockIdx.x * L_;   // blockIdx.x in [0, B*H)
    int tid = threadIdx.x;                          // 256 threads
    __shared__ float red[256];

    float m = -3.402823e38f;
    for (int i = tid; i < L_; i += 256) m = fmaxf(m, s[i]);
    red[tid] = m; __syncthreads();
    for (int off = 128; off > 0; off >>= 1) {
        if (tid < off) red[tid] = fmaxf(red[tid], red[tid + off]);
        __syncthreads();
    }
    float mx = red[0];
    __syncthreads();

    float sum = 0.f;
    for (int i = tid; i < L_; i += 256) {
        float e = __expf(s[i] - mx);
        s[i] = e;
        sum += e;
    }
    red[tid] = sum; __syncthreads();
    for (int off = 128; off > 0; off >>= 1) {
        if (tid < off) red[tid] += red[tid + off];
        __syncthreads();
    }
    float inv = 1.f / red[0];
    for (int i = tid; i < L_; i += 256) s[i] *= inv;
}

// ---------------------------------------------------------------------------
// Phase 4: ctx[b,h,c] = sum_l probs[b,h,l] * kv[b,l,c]  (WMMA, split-K over L,
// reduced with global_atomic_add_f32). ctx must be zeroed first.
// ---------------------------------------------------------------------------
__global__ void zero_kernel(float* __restrict__ p, int n) {
    int i = blockIdx.x * blockDim.x + threadIdx.x;
    if (i < n) p[i] = 0.f;
}

__global__ void ctx_kernel(const float* __restrict__ probs, const float* __restrict__ kv,
                           float* __restrict__ ctx) {
    int wave   = (blockIdx.x * blockDim.x + threadIdx.x) >> 5;  // 0..2047
    int lane   = threadIdx.x & 31;
    int kchunk = wave & 7;            // 8 chunks of 512 l's
    int cbase  = ((wave >> 3) & 31) * 16;
    int b      = wave >> 8;
    int m      = lane & 15;
    int khalf  = lane >> 4;

    const float* a0row = probs + (size_t)(b * H_ + m)      * L_;
    const float* a1row = probs + (size_t)(b * H_ + 16 + m) * L_;
    const float* kvb   = kv + (size_t)b * L_ * C_;

    v8f d0 = {}; v8f d1 = {};
    int k0 = kchunk * 512;
    for (int kb = k0; kb < k0 + 512; kb += 4) {
        int kcol = kb + 2 * khalf;
        v2f a0 = *(const v2f*)(a0row + kcol);
        v2f a1 = *(const v2f*)(a1row + kcol);
        v2f bb;                       // B[k=l][n=c]: two rows of kv, column cbase+m
        bb.x = kvb[(size_t)kcol       * C_ + cbase + m];
        bb.y = kvb[(size_t)(kcol + 1) * C_ + cbase + m];
        d0 = __builtin_amdgcn_wmma_f32_16x16x4_f32(false, a0, false, bb, (short)0, d0, false, false);
        d1 = __builtin_amdgcn_wmma_f32_16x16x4_f32(false, a1, false, bb, (short)0, d1, false, false);
    }

    float* crow = ctx + (size_t)(b * H_) * C_ + cbase + m;
#pragma unroll
    for (int r = 0; r < 8; ++r) {
        int row0 = khalf * 8 + r;
        atomicAdd(crow + (size_t)row0        * C_, d0[r]);
        atomicAdd(crow + (size_t)(row0 + 16) * C_, d1[r]);
    }
}

// ---------------------------------------------------------------------------
// Phase 5: out[b, h*K+k] = sum_c ctx[b,h,c] * v_w[h*K+k, c]
// ---------------------------------------------------------------------------
__global__ void out_kernel(const float* __restrict__ ctx, const float* __restrict__ vw,
                           float* __restrict__ out) {
    int idx = blockIdx.x * blockDim.x + threadIdx.x;   // 0 .. B*HK-1
    int b  = idx >> 12;
    int hk = idx & (HK_ - 1);
    int h  = hk >> 7;
    const float* crow = ctx + (size_t)(b * H_ + h) * C_;
    const float* wrow = vw  + (size_t)hk * C_;
    float acc = 0.f;
#pragma unroll 4
    for (int c = 0; c < C_; c += 4) {
        v4f cv = *(const v4f*)(crow + c);
        v4f wv = *(const v4f*)(wrow + c);
        acc += cv.x * wv.x + cv.y * wv.y + cv.z * wv.z + cv.w * wv.w;
    }
    out[idx] = acc;
}

// ---------------------------------------------------------------------------
extern "C" void kernel_launch(void* const* d_in, const int* in_sizes, int n_in,
                              void* d_out, int out_size, void* d_ws, size_t ws_size,
                              hipStream_t stream) {
    const float* hq = (const float*)d_in[0];   // [B, QC]
    const float* kv = (const float*)d_in[1];   // [B, L, C]
    const float* qw = (const float*)d_in[2];   // [HK, QC]
    const float* qb = (const float*)d_in[3];   // [HK]
    const float* kw = (const float*)d_in[4];   // [HK, C]
    const float* vw = (const float*)d_in[5];   // [HK, C]
    float* out = (float*)d_out;                // [B, HK]

    float* ws     = (float*)d_ws;
    float* q      = ws;                               // B*HK      = 32768
    float* qeff   = q      + (size_t)B_ * HK_;        // B*H*C     = 131072
    float* scores = qeff   + (size_t)B_ * H_ * C_;    // B*H*L     = 1048576
    float* ctx    = scores + (size_t)B_ * H_ * L_;    // B*H*C     = 131072

    qproj_kernel  <<<(B_ * HK_) / 256, 256, 0, stream>>>(hq, qw, qb, q);
    qeff_kernel   <<<(B_ * H_ * C_) / 256, 256, 0, stream>>>(q, kw, qeff);
    scores_kernel <<<(B_ * (L_ / 16) * 32) / 256, 256, 0, stream>>>(qeff, kv, scores);
    softmax_kernel<<<B_ * H_, 256, 0, stream>>>(scores);
    zero_kernel   <<<(B_ * H_ * C_) / 256, 256, 0, stream>>>(ctx, B_ * H_ * C_);
    ctx_kernel    <<<(B_ * 32 * 8 * 32) / 256, 256, 0, stream>>>(scores, kv, ctx);
    out_kernel    <<<(B_ * HK_) / 256, 256, 0, stream>>>(ctx, vw, out);
}